// ST_HGNN_Layer_72859825209390
// MI455X (gfx1250) — compile-verified
//
#include <hip/hip_runtime.h>
#include <hip/hip_bf16.h>

typedef __attribute__((ext_vector_type(2))) float v2f;
typedef __attribute__((ext_vector_type(8))) float v8f;

#define BMT 32    // B*M batches
#define NN  1024  // nodes
#define DD  128   // input dim
#define PP  64    // projection dim
#define NC  10    // clusters
#define FF  128   // output features
#define KNN 10    // k neighbors
#define NE  (NN + NC)  // 1034 hyperedges

static __device__ __forceinline__ v8f wmma_f32(v2f a, v2f b, v8f c) {
  // D = A(16x4 f32) x B(4x16 f32) + C(16x16 f32)
  return __builtin_amdgcn_wmma_f32_16x16x4_f32(
      /*neg_a=*/false, a, /*neg_b=*/false, b,
      /*c_mod=*/(short)0, c, /*reuse_a=*/false, /*reuse_b=*/false);
}

// ---------------------------------------------------------------------------
// Out[bm][n][j] = sum_d x_flat[bm][n][d] * W[j][d] + bias[j]
// x_flat[bm][n][d] = x[b][n][m][d], bm = b*8 + m  (transpose folded into addr)
// One wave computes one 16x16 tile; K=128 via 32 chained f32 WMMAs.
// ---------------------------------------------------------------------------
__global__ void __launch_bounds__(32)
gemm_xw_kernel(const float* __restrict__ x, const float* __restrict__ W,
               const float* __restrict__ bias, float* __restrict__ out, int ncols) {
  const int lane = threadIdx.x;
  const int jt = blockIdx.x, nt = blockIdx.y, bm = blockIdx.z;
  const int b = bm >> 3, m = bm & 7;
  const int half = lane >> 4, l16 = lane & 15;
  const int arow = nt * 16 + l16;   // n (A-matrix M index per lane)
  const int bcol = jt * 16 + l16;   // j (B-matrix N index per lane)
  const float* __restrict__ ax = x + ((((size_t)b * NN + arow) * 8 + m) * DD);
  const float* __restrict__ wj = W + (size_t)bcol * DD;
  v8f acc = {0.f, 0.f, 0.f, 0.f, 0.f, 0.f, 0.f, 0.f};
#pragma unroll
  for (int k = 0; k < DD; k += 4) {
    const int ka = k + half * 2;       // lanes 16-31 carry K=2,3 of the step
    v2f av; av.x = ax[ka]; av.y = ax[ka + 1];
    v2f bv; bv.x = wj[ka]; bv.y = wj[ka + 1];
    acc = wmma_f32(av, bv, acc);
  }
  const float bj = bias[bcol];
  float* __restrict__ op = out + (size_t)bm * NN * ncols;
#pragma unroll
  for (int r = 0; r < 8; ++r) {
    const int row = nt * 16 + r + 8 * half;  // C/D layout: VGPR r -> M = r (+8)
    op[(size_t)row * ncols + bcol] = acc[r] + bj;
  }
}

// sq[bm*N + n] = ||Z[bm][n][:]||^2
__global__ void sq_kernel(const float* __restrict__ Z, float* __restrict__ sq) {
  const int gid = blockIdx.x * blockDim.x + threadIdx.x;
  if (gid >= BMT * NN) return;
  const float* __restrict__ z = Z + (size_t)gid * PP;
  float s = 0.f;
#pragma unroll
  for (int p = 0; p < PP; ++p) s += z[p] * z[p];
  sq[gid] = s;
}

// ---------------------------------------------------------------------------
// Fused Gram + top-k: per (bm, 16-row tile) compute d2 to all 1024 columns via
// WMMA (8 waves x 8 column tiles each), stage the 16x1024 slab in LDS (64KB),
// then a two-phase parallel top-10:
//   phase 1: 16 threads per row, each keeps a sorted top-10 of its 64 columns
//   phase 2: spill partial lists to LDS (slab reused), 1 thread/row merges 16
//            sorted lists with early exit. Stable: strict '<' => lower index
//            wins ties, matching lax.top_k.
// d2 never hits HBM (saves 134 MB of traffic).
// ---------------------------------------------------------------------------
__global__ void __launch_bounds__(256)
gram_topk_kernel(const float* __restrict__ Z, const float* __restrict__ sq,
                 int* __restrict__ idx) {
  __shared__ float dist[16 * NN];  // 64 KB; reused as merge scratch
  const int nt = blockIdx.x, bm = blockIdx.y;
  const int tid = threadIdx.x;
  const int wave = tid >> 5, lane = tid & 31;
  const int half = lane >> 4, l16 = lane & 15;
  const float* __restrict__ Zb  = Z  + (size_t)bm * NN * PP;
  const float* __restrict__ sqb = sq + (size_t)bm * NN;
  const float* __restrict__ za  = Zb + (size_t)(nt * 16 + l16) * PP;

  for (int ct = wave; ct < NN / 16; ct += 8) {   // wave-uniform bounds
    const int j = ct * 16 + l16;
    const float* __restrict__ zb = Zb + (size_t)j * PP;
    v8f acc = {0.f, 0.f, 0.f, 0.f, 0.f, 0.f, 0.f, 0.f};
#pragma unroll
    for (int k = 0; k < PP; k += 4) {
      const int ka = k + half * 2;
      v2f av; av.x = za[ka]; av.y = za[ka + 1];
      v2f bv; bv.x = zb[ka]; bv.y = zb[ka + 1];
      acc = wmma_f32(av, bv, acc);
    }
    const float sj = sqb[j];
#pragma unroll
    for (int r = 0; r < 8; ++r) {
      const int mrow = r + 8 * half;
      dist[mrow * NN + j] = sqb[nt * 16 + mrow] + sj - 2.f * acc[r];
    }
  }
  __syncthreads();

  // --- phase 1: per-thread partial top-10 over a 64-column slice ---
  const int prow = tid >> 4;      // row 0..15
  const int pcol = tid & 15;      // slice 0..15
  float bd[KNN]; int bi[KNN];
#pragma unroll
  for (int t = 0; t < KNN; ++t) { bd[t] = 3.4e38f; bi[t] = 0; }
  {
    const float* __restrict__ dr = &dist[prow * NN + pcol * 64];
    const int jbase = pcol * 64;
    for (int j = 0; j < 64; ++j) {
      const float d = dr[j];
      if (d < bd[KNN - 1]) {
        bd[KNN - 1] = d; bi[KNN - 1] = jbase + j;
#pragma unroll
        for (int t = KNN - 1; t > 0; --t) {
          if (bd[t] < bd[t - 1]) {
            float td = bd[t]; bd[t] = bd[t - 1]; bd[t - 1] = td;
            int   ti = bi[t]; bi[t] = bi[t - 1]; bi[t - 1] = ti;
          }
        }
      }
    }
  }
  __syncthreads();  // all reads of the distance slab done; reuse as scratch

  // spill partial lists: row r, slice t at entries (r*16+t)*20 .. +19
  {
    float* __restrict__ sf = &dist[(prow * 16 + pcol) * 20];
    int*   __restrict__ si = (int*)sf + KNN;
#pragma unroll
    for (int t = 0; t < KNN; ++t) { sf[t] = bd[t]; si[t] = bi[t]; }
  }
  __syncthreads();

  // --- phase 2: one thread per row merges 16 sorted lists of 10 ---
  if (tid < 16) {
    float md[KNN]; int mi[KNN];
#pragma unroll
    for (int t = 0; t < KNN; ++t) { md[t] = 3.4e38f; mi[t] = 0; }
    for (int s = 0; s < 16; ++s) {
      const float* __restrict__ sf = &dist[(tid * 16 + s) * 20];
      const int*   __restrict__ si = (const int*)sf + KNN;
      for (int k = 0; k < KNN; ++k) {
        const float d = sf[k];
        if (d >= md[KNN - 1]) break;   // list sorted ascending: rest can't enter
        md[KNN - 1] = d; mi[KNN - 1] = si[k];
#pragma unroll
        for (int t = KNN - 1; t > 0; --t) {
          if (md[t] < md[t - 1]) {
            float td = md[t]; md[t] = md[t - 1]; md[t - 1] = td;
            int   ti = mi[t]; mi[t] = mi[t - 1]; mi[t - 1] = ti;
          }
        }
      }
    }
    int* __restrict__ op = idx + ((size_t)bm * NN + nt * 16 + tid) * KNN;
#pragma unroll
    for (int t = 0; t < KNN; ++t) op[t] = mi[t];
  }
}

// Hc = softmax(Z @ C^T); Dv = K + sum(Hc row); De[cluster cols] += column sums
__global__ void cluster_softmax_kernel(const float* __restrict__ Z,
                                       const float* __restrict__ C,
                                       float* __restrict__ Hc,
                                       float* __restrict__ Dv,
                                       float* __restrict__ De) {
  const int gid = blockIdx.x * blockDim.x + threadIdx.x;
  if (gid >= BMT * NN) return;
  const int bm = gid >> 10;
  const float* __restrict__ z = Z + (size_t)gid * PP;
  float logit[NC];
  float mx = -3.4e38f;
#pragma unroll
  for (int c = 0; c < NC; ++c) {
    float s = 0.f;
#pragma unroll
    for (int p = 0; p < PP; ++p) s += z[p] * C[c * PP + p];
    logit[c] = s;
    mx = fmaxf(mx, s);
  }
  float den = 0.f;
#pragma unroll
  for (int c = 0; c < NC; ++c) { logit[c] = expf(logit[c] - mx); den += logit[c]; }
  const float inv = 1.f / den;
  float sum = 0.f;
#pragma unroll
  for (int c = 0; c < NC; ++c) {
    const float h = logit[c] * inv;
    Hc[(size_t)gid * NC + c] = h;
    sum += h;
    atomicAdd(&De[(size_t)bm * NE + NN + c], h);
  }
  Dv[gid] = (float)KNN + sum;  // knn row-sum is exactly K ones
}

// De[knn cols]: occurrence counts of each neighbor index
__global__ void knn_degree_kernel(const int* __restrict__ idx, float* __restrict__ De) {
  const int gid = blockIdx.x * blockDim.x + threadIdx.x;
  if (gid >= BMT * NN * KNN) return;
  const int bm = gid / (NN * KNN);
  atomicAdd(&De[(size_t)bm * NE + idx[gid]], 1.0f);
}

// T[e][f] += H_knn^T scatter of Y = Dv^-1/2 * X_trans
__global__ void scatter_T_kernel(const float* __restrict__ Xt, const float* __restrict__ Dv,
                                 const int* __restrict__ idx, float* __restrict__ T) {
  const int gid = blockIdx.x * blockDim.x + threadIdx.x;
  if (gid >= BMT * NN * FF) return;
  const int f = gid & (FF - 1);
  const int row = gid >> 7;       // bm*N + n
  const int bm = row >> 10;
  const float y = rsqrtf(fmaxf(Dv[row], 1e-12f)) * Xt[gid];
  const int* __restrict__ id = idx + (size_t)row * KNN;
  float* __restrict__ Tb = T + (size_t)bm * NE * FF;
#pragma unroll
  for (int k = 0; k < KNN; ++k)
    atomicAdd(&Tb[(size_t)id[k] * FF + f], y);
}

// T[N+c][f] = sum_n Hc[n][c] * Dv^-1/2[n] * Xt[n][f]  (dense cluster rows)
__global__ void cluster_T_kernel(const float* __restrict__ Hc, const float* __restrict__ Xt,
                                 const float* __restrict__ Dv, float* __restrict__ T) {
  const int c = blockIdx.x, bm = blockIdx.y, f = threadIdx.x;
  float s = 0.f;
  for (int n = 0; n < NN; ++n) {
    const int row = bm * NN + n;
    s += Hc[(size_t)row * NC + c] * rsqrtf(fmaxf(Dv[row], 1e-12f)) *
         Xt[(size_t)row * FF + f];
  }
  T[((size_t)bm * NE + NN + c) * FF + f] = s;
}

// T[e][f] *= De_inv[e]
__global__ void norm_T_kernel(float* __restrict__ T, const float* __restrict__ De) {
  const int gid = blockIdx.x * blockDim.x + threadIdx.x;
  if (gid >= BMT * NE * FF) return;
  const int row = gid >> 7;
  const float de = De[row];
  const float inv = (de > 0.f) ? (1.f / fmaxf(de, 1e-12f)) : 0.f;
  T[gid] *= inv;
}

// out = elu(Dv^-1/2 * (gather_knn(T) + Hc @ T_cluster)), written as [B,N,M,F]
__global__ void gather_out_kernel(const float* __restrict__ T, const int* __restrict__ idx,
                                  const float* __restrict__ Hc, const float* __restrict__ Dv,
                                  float* __restrict__ out) {
  const int gid = blockIdx.x * blockDim.x + threadIdx.x;
  if (gid >= BMT * NN * FF) return;
  const int f = gid & (FF - 1);
  const int row = gid >> 7;
  const int n = row & (NN - 1);
  const int bm = row >> 10;
  const float* __restrict__ Tb = T + (size_t)bm * NE * FF;
  const int* __restrict__ id = idx + (size_t)row * KNN;
  const float* __restrict__ hc = Hc + (size_t)row * NC;
  float s = 0.f;
#pragma unroll
  for (int k = 0; k < KNN; ++k) s += Tb[(size_t)id[k] * FF + f];
#pragma unroll
  for (int c = 0; c < NC; ++c) s += hc[c] * Tb[(size_t)(NN + c) * FF + f];
  s *= rsqrtf(fmaxf(Dv[row], 1e-12f));
  const float r = (s > 0.f) ? s : expm1f(s);
  const int b = bm >> 3, m = bm & 7;
  out[(((size_t)b * NN + n) * 8 + m) * FF + f] = r;
}

extern "C" void kernel_launch(void* const* d_in, const int* in_sizes, int n_in,
                              void* d_out, int out_size, void* d_ws, size_t ws_size,
                              hipStream_t stream) {
  (void)in_sizes; (void)n_in; (void)out_size; (void)ws_size;
  const float* x  = (const float*)d_in[0];  // [4,1024,8,128]
  const float* Wp = (const float*)d_in[1];  // [64,128]
  const float* bp = (const float*)d_in[2];  // [64]
  const float* C  = (const float*)d_in[3];  // [10,64]
  const float* Wt = (const float*)d_in[4];  // [128,128]
  const float* bt = (const float*)d_in[5];  // [128]
  float* out = (float*)d_out;

  // Workspace layout (floats), total ~45 MB
  float* Z  = (float*)d_ws;                   // 32*1024*64
  float* Xt = Z  + (size_t)BMT * NN * PP;     // 32*1024*128
  float* sq = Xt + (size_t)BMT * NN * FF;     // 32*1024
  float* Dv = sq + (size_t)BMT * NN;          // 32*1024
  float* Hc = Dv + (size_t)BMT * NN;          // 32*1024*10
  float* De = Hc + (size_t)BMT * NN * NC;     // 32*1034
  float* T  = De + (size_t)BMT * NE;          // 32*1034*128
  int* idx  = (int*)(T + (size_t)BMT * NE * FF);  // 32*1024*10

  // zero De and T (contiguous) before the atomic accumulations
  hipMemsetAsync(De, 0, (size_t)(BMT * NE + (size_t)BMT * NE * FF) * sizeof(float), stream);

  gemm_xw_kernel<<<dim3(PP / 16, NN / 16, BMT), 32, 0, stream>>>(x, Wp, bp, Z, PP);
  gemm_xw_kernel<<<dim3(FF / 16, NN / 16, BMT), 32, 0, stream>>>(x, Wt, bt, Xt, FF);
  sq_kernel<<<(BMT * NN) / 256, 256, 0, stream>>>(Z, sq);
  gram_topk_kernel<<<dim3(NN / 16, BMT), 256, 0, stream>>>(Z, sq, idx);
  cluster_softmax_kernel<<<(BMT * NN) / 256, 256, 0, stream>>>(Z, C, Hc, Dv, De);
  knn_degree_kernel<<<(BMT * NN * KNN + 255) / 256, 256, 0, stream>>>(idx, De);
  scatter_T_kernel<<<(BMT * NN * FF) / 256, 256, 0, stream>>>(Xt, Dv, idx, T);
  cluster_T_kernel<<<dim3(NC, BMT), FF, 0, stream>>>(Hc, Xt, Dv, T);
  norm_T_kernel<<<(BMT * NE * FF + 255) / 256, 256, 0, stream>>>(T, De);
  gather_out_kernel<<<(BMT * NN * FF) / 256, 256, 0, stream>>>(T, idx, Hc, Dv, out);
}